// GKANLayer_87789131531089
// MI455X (gfx1250) — compile-verified
//
#include <hip/hip_runtime.h>
#include <math.h>

// ---------------------------------------------------------------------------
// GKAN layer for MI455X (gfx1250, wave32, WMMA)
// B=8, N=2048, D_IN=D_OUT=128, K_NBR=8
// ---------------------------------------------------------------------------

typedef __attribute__((ext_vector_type(2))) float v2f;
typedef __attribute__((ext_vector_type(8))) float v8f;

#define B_    8
#define N_    2048
#define D_    128
#define KNBR  8
#define ROWS  (B_ * N_)          // 16384
#define BIGF  3.0e38f

// D = A(16x4) * B(4x16) + C(16x16), fp32, one wave.
__device__ __forceinline__ v8f wmma_f32(v2f a, v2f b, v8f c) {
  return __builtin_amdgcn_wmma_f32_16x16x4_f32(false, a, false, b, (short)0, c,
                                               false, false);
}

// ---------------------------------------------------------------------------
// Kernel A0: squared norms xx[row] = sum_k x[row][k]^2   (one wave per row)
// ---------------------------------------------------------------------------
__global__ __launch_bounds__(256) void k_xx(const float* __restrict__ x,
                                            float* __restrict__ xx) {
  int wid  = (blockIdx.x * blockDim.x + threadIdx.x) >> 5;  // row
  int lane = threadIdx.x & 31;
  if (wid >= ROWS) return;  // uniform per wave
  const float* r = x + (size_t)wid * D_;
  float s = 0.0f;
  #pragma unroll
  for (int i = 0; i < D_ / 32; ++i) {
    float v = r[lane + 32 * i];
    s += v * v;
  }
  #pragma unroll
  for (int m = 16; m >= 1; m >>= 1) s += __shfl_xor(s, m, 32);
  if (lane == 0) xx[wid] = s;
}

// ---------------------------------------------------------------------------
// Kernel A: per-batch pairwise distances via WMMA Gram tiles + top-9 select.
// Grid: (B*N/16) workgroups of 512 threads (16 waves).
// Workgroup = one 16-row strip of one batch. Wave w handles column tiles
// ct = w + 16*t, t=0..7 (covers all 128 tiles of N=2048 columns).
// Dual accumulators break the WMMA RAW chain (dep distance 2).
// ---------------------------------------------------------------------------
__global__ __launch_bounds__(512) void k_knn(const float* __restrict__ x,
                                             const float* __restrict__ xx,
                                             int* __restrict__ nbr) {
  __shared__ float stage[16][16][17];   // [wave][m][n] (+pad)
  __shared__ float cval[16][16 * 9];    // [row][wave*9 + k]
  __shared__ int   cidx[16][16 * 9];

  const int bid   = blockIdx.x;
  const int b     = bid >> 7;           // batch
  const int strip = bid & 127;          // 16-row strip within batch
  const int tid   = threadIdx.x;
  const int wave  = tid >> 5;
  const int lane  = tid & 31;
  const int half  = lane >> 4;
  const int l16   = lane & 15;

  const int rowBase = b * N_ + strip * 16;     // global row of strip start

  // Preload full-K A fragments for this strip (row = l16 of strip).
  const float* xa = x + (size_t)(rowBase + l16) * D_;
  v2f areg[32];
  #pragma unroll
  for (int kk = 0; kk < 32; ++kk)
    areg[kk] = *(const v2f*)(xa + 4 * kk + 2 * half);

  // Row squared-norms seen by this lane's C fragment (rows v + 8*half).
  float xxr[8];
  #pragma unroll
  for (int v = 0; v < 8; ++v) xxr[v] = xx[rowBase + v + 8 * half];

  // Per-row top-9 (lanes 0..15 own row = lane).
  float bv[9];
  int   bi[9];
  #pragma unroll
  for (int k = 0; k < 9; ++k) { bv[k] = BIGF; bi[k] = 0x7fffffff; }

  for (int t = 0; t < 8; ++t) {
    const int ct = wave + 16 * t;       // column tile index
    const int c0 = ct * 16;
    const float* xb = x + (size_t)(b * N_ + c0 + l16) * D_;
    const float  xxc = xx[b * N_ + c0 + l16];

    v8f acc0 = {0.f, 0.f, 0.f, 0.f, 0.f, 0.f, 0.f, 0.f};
    v8f acc1 = {0.f, 0.f, 0.f, 0.f, 0.f, 0.f, 0.f, 0.f};
    #pragma unroll
    for (int kk = 0; kk < 32; kk += 2) {
      v2f b0 = *(const v2f*)(xb + 4 * kk + 2 * half);
      v2f b1 = *(const v2f*)(xb + 4 * (kk + 1) + 2 * half);
      acc0 = wmma_f32(areg[kk],     b0, acc0);
      acc1 = wmma_f32(areg[kk + 1], b1, acc1);
    }

    // dist = ||a||^2 - 2 a.b + ||b||^2, clipped at 0; stage to LDS.
    #pragma unroll
    for (int v = 0; v < 8; ++v) {
      float d = xxr[v] + xxc - 2.0f * (acc0[v] + acc1[v]);
      stage[wave][v + 8 * half][l16] = fmaxf(d, 0.0f);
    }
    __syncthreads();

    if (lane < 16) {                    // lane owns row m = lane
      const int m = lane;
      #pragma unroll
      for (int n = 0; n < 16; ++n) {
        float d  = stage[wave][m][n];
        int   gi = c0 + n;              // column index within batch
        if (d < bv[8] || (d == bv[8] && gi < bi[8])) {
          bv[8] = d; bi[8] = gi;
          #pragma unroll
          for (int j = 8; j >= 1; --j) {
            bool sw = (bv[j] < bv[j - 1]) ||
                      (bv[j] == bv[j - 1] && bi[j] < bi[j - 1]);
            if (sw) {
              float tv = bv[j]; bv[j] = bv[j - 1]; bv[j - 1] = tv;
              int   ti = bi[j]; bi[j] = bi[j - 1]; bi[j - 1] = ti;
            }
          }
        }
      }
    }
    __syncthreads();
  }

  // Dump per-wave candidates.
  if (lane < 16) {
    #pragma unroll
    for (int k = 0; k < 9; ++k) {
      cval[lane][wave * 9 + k] = bv[k];
      cidx[lane][wave * 9 + k] = bi[k];
    }
  }
  __syncthreads();

  // Merge: wave w extracts the 9 smallest of row w's 144 candidates.
  {
    const int row = wave;
    float* CV = cval[row];
    int*   CI = cidx[row];
    for (int k = 0; k < 9; ++k) {
      float bval = BIGF; int bgi = 0x7fffffff; int bpos = -1;
      for (int p = lane; p < 144; p += 32) {
        float v = CV[p]; int gi = CI[p];
        if (v < bval || (v == bval && gi < bgi)) { bval = v; bgi = gi; bpos = p; }
      }
      #pragma unroll
      for (int m = 16; m >= 1; m >>= 1) {
        float v2 = __shfl_xor(bval, m, 32);
        int   g2 = __shfl_xor(bgi, m, 32);
        int   p2 = __shfl_xor(bpos, m, 32);
        if (v2 < bval || (v2 == bval && g2 < bgi)) { bval = v2; bgi = g2; bpos = p2; }
      }
      // Entry 0 is self (dist 0); entries 1..8 are the neighbors.
      if (k >= 1 && lane == 0)
        nbr[(size_t)(rowBase + row) * KNBR + (k - 1)] = b * N_ + bgi; // global row
      if (bpos >= 0 && (bpos & 31) == lane) CV[bpos] = BIGF;
    }
  }
}

// ---------------------------------------------------------------------------
// Kernel B: z = x @ W_nbr,  hp = x @ W_self + z + bias
// Grid: ROWS/16 workgroups of 256 threads (8 waves); wave w -> col tile w.
// Four accumulators (2 per output) give WMMA dependency distance 4.
// ---------------------------------------------------------------------------
__global__ __launch_bounds__(256) void k_gemm(const float* __restrict__ x,
                                              const float* __restrict__ Wself,
                                              const float* __restrict__ Wnbr,
                                              const float* __restrict__ bias,
                                              float* __restrict__ zf,
                                              float* __restrict__ hp) {
  const int r0   = blockIdx.x * 16;
  const int tid  = threadIdx.x;
  const int wave = tid >> 5;
  const int lane = tid & 31;
  const int half = lane >> 4;
  const int l16  = lane & 15;
  const int c0   = wave * 16;

  const float* xa = x + (size_t)(r0 + l16) * D_;
  v2f areg[32];
  #pragma unroll
  for (int kk = 0; kk < 32; ++kk)
    areg[kk] = *(const v2f*)(xa + 4 * kk + 2 * half);

  v8f accS0 = {0.f, 0.f, 0.f, 0.f, 0.f, 0.f, 0.f, 0.f};
  v8f accS1 = {0.f, 0.f, 0.f, 0.f, 0.f, 0.f, 0.f, 0.f};
  v8f accZ0 = {0.f, 0.f, 0.f, 0.f, 0.f, 0.f, 0.f, 0.f};
  v8f accZ1 = {0.f, 0.f, 0.f, 0.f, 0.f, 0.f, 0.f, 0.f};
  #pragma unroll
  for (int kk = 0; kk < 32; kk += 2) {
    const int k0 = 4 * kk + 2 * half;
    const int k1 = 4 * (kk + 1) + 2 * half;
    v2f bs0, bz0, bs1, bz1;
    bs0[0] = Wself[(size_t)(k0 + 0) * D_ + c0 + l16];
    bs0[1] = Wself[(size_t)(k0 + 1) * D_ + c0 + l16];
    bz0[0] = Wnbr [(size_t)(k0 + 0) * D_ + c0 + l16];
    bz0[1] = Wnbr [(size_t)(k0 + 1) * D_ + c0 + l16];
    bs1[0] = Wself[(size_t)(k1 + 0) * D_ + c0 + l16];
    bs1[1] = Wself[(size_t)(k1 + 1) * D_ + c0 + l16];
    bz1[0] = Wnbr [(size_t)(k1 + 0) * D_ + c0 + l16];
    bz1[1] = Wnbr [(size_t)(k1 + 1) * D_ + c0 + l16];
    accS0 = wmma_f32(areg[kk],     bs0, accS0);
    accZ0 = wmma_f32(areg[kk],     bz0, accZ0);
    accS1 = wmma_f32(areg[kk + 1], bs1, accS1);
    accZ1 = wmma_f32(areg[kk + 1], bz1, accZ1);
  }

  const float bvv = bias[c0 + l16];
  #pragma unroll
  for (int v = 0; v < 8; ++v) {
    const size_t r  = (size_t)(r0 + v + 8 * half);
    const float  zv = accZ0[v] + accZ1[v];
    zf[r * D_ + c0 + l16] = zv;
    hp[r * D_ + c0 + l16] = accS0[v] + accS1[v] + zv + bvv;
  }
}

// ---------------------------------------------------------------------------
// Kernel C: neighbor aggregation + KAN rational activation.
// ---------------------------------------------------------------------------
__global__ __launch_bounds__(256) void k_agg(const float* __restrict__ zf,
                                             const float* __restrict__ hp,
                                             const int* __restrict__ nbr,
                                             const float* __restrict__ kan_a,
                                             const float* __restrict__ kan_b,
                                             float* __restrict__ hact) {
  const int idx = blockIdx.x * 256 + threadIdx.x;   // 0 .. ROWS*D_-1
  const int row = idx >> 7;
  const int f   = idx & 127;

  const int* nb = nbr + (size_t)row * KNBR;
  float acc = 0.0f;
  #pragma unroll
  for (int j = 0; j < KNBR; ++j) acc += zf[(size_t)nb[j] * D_ + f];

  const float scale = 1.0f / (8.0f + 1e-6f);  // deg_inv_sqrt[n]*deg_inv_sqrt[m]
  float h  = hp[idx] + acc * scale;
  float h2 = h * h;
  float num = kan_a[f * 3 + 0] + kan_a[f * 3 + 1] * h + kan_a[f * 3 + 2] * h2;
  float den = 1.0f + fabsf(kan_b[f * 2 + 0] * h + kan_b[f * 2 + 1] * h2);
  hact[idx] = num / (den + 1e-8f);
}

// ---------------------------------------------------------------------------
// Kernel D1: per-chunk partial sums (coalesced). 64 blocks x 256 threads;
// block g reduces rows [g*256, (g+1)*256) into partial sum/sumsq per feature.
// ---------------------------------------------------------------------------
__global__ __launch_bounds__(256) void k_stats1(const float* __restrict__ hact,
                                                float* __restrict__ psum,
                                                float* __restrict__ psq) {
  const int g  = blockIdx.x;
  const int f  = threadIdx.x & 127;
  const int rh = threadIdx.x >> 7;       // 0 or 1
  const int base = g * 256;
  float a = 0.0f, q = 0.0f;
  for (int i = 0; i < 128; ++i) {
    float v = hact[(size_t)(base + 2 * i + rh) * D_ + f];
    a += v;
    q += v * v;
  }
  __shared__ float s1[256];
  __shared__ float s2[256];
  s1[threadIdx.x] = a;
  s2[threadIdx.x] = q;
  __syncthreads();
  if (rh == 0) {
    psum[(size_t)g * D_ + f] = s1[f] + s1[128 + f];
    psq [(size_t)g * D_ + f] = s2[f] + s2[128 + f];
  }
}

// ---------------------------------------------------------------------------
// Kernel D2: fold 64 partials per feature -> mean/var. 1 block x 128 threads.
// ---------------------------------------------------------------------------
__global__ __launch_bounds__(128) void k_stats2(const float* __restrict__ psum,
                                                const float* __restrict__ psq,
                                                float* __restrict__ meanvar) {
  const int f = threadIdx.x;
  float a = 0.0f, q = 0.0f;
  for (int g = 0; g < 64; ++g) {
    a += psum[(size_t)g * D_ + f];
    q += psq [(size_t)g * D_ + f];
  }
  float m   = a * (1.0f / ROWS);
  float var = q * (1.0f / ROWS) - m * m;
  meanvar[f]       = m;
  meanvar[128 + f] = var;
}

// ---------------------------------------------------------------------------
// Kernel E: batch-norm affine.
// ---------------------------------------------------------------------------
__global__ __launch_bounds__(256) void k_bn(const float* __restrict__ hact,
                                            const float* __restrict__ meanvar,
                                            const float* __restrict__ gamma,
                                            const float* __restrict__ beta,
                                            float* __restrict__ out) {
  const int idx = blockIdx.x * 256 + threadIdx.x;
  const int f   = idx & 127;
  float m = meanvar[f], v = meanvar[128 + f];
  out[idx] = (hact[idx] - m) * rsqrtf(v + 1e-5f) * gamma[f] + beta[f];
}

// ---------------------------------------------------------------------------
extern "C" void kernel_launch(void* const* d_in, const int* in_sizes, int n_in,
                              void* d_out, int out_size, void* d_ws,
                              size_t ws_size, hipStream_t stream) {
  const float* x     = (const float*)d_in[0];
  const float* Wself = (const float*)d_in[1];
  const float* Wnbr  = (const float*)d_in[2];
  const float* kan_a = (const float*)d_in[3];
  const float* kan_b = (const float*)d_in[4];
  const float* bias  = (const float*)d_in[5];
  const float* gamma = (const float*)d_in[6];
  const float* beta  = (const float*)d_in[7];

  float* ws      = (float*)d_ws;
  float* xx      = ws;                          // 16384 f32
  int*   nbr     = (int*)(ws + ROWS);           // 16384*8 i32
  float* zf      = ws + ROWS + ROWS * KNBR;     // 16384*128 f32
  float* hp      = zf + (size_t)ROWS * D_;      // 16384*128 f32
  float* hact    = hp + (size_t)ROWS * D_;      // 16384*128 f32
  float* psum    = hact + (size_t)ROWS * D_;    // 64*128 f32
  float* psq     = psum + 64 * D_;              // 64*128 f32
  float* meanvar = psq + 64 * D_;               // 256 f32

  k_xx    <<<ROWS / 8,        256, 0, stream>>>(x, xx);
  k_knn   <<<ROWS / 16,       512, 0, stream>>>(x, xx, nbr);
  k_gemm  <<<ROWS / 16,       256, 0, stream>>>(x, Wself, Wnbr, bias, zf, hp);
  k_agg   <<<ROWS * D_ / 256, 256, 0, stream>>>(zf, hp, nbr, kan_a, kan_b, hact);
  k_stats1<<<64,              256, 0, stream>>>(hact, psum, psq);
  k_stats2<<<1,               128, 0, stream>>>(psum, psq, meanvar);
  k_bn    <<<ROWS * D_ / 256, 256, 0, stream>>>(hact, meanvar, gamma, beta,
                                                (float*)d_out);
}